// MFA_42021960024360
// MI455X (gfx1250) — compile-verified
//
#include <hip/hip_runtime.h>
#include <hip/hip_bf16.h>
#include <math.h>

typedef __attribute__((ext_vector_type(2))) float v2f;
typedef __attribute__((ext_vector_type(8))) float v8f;

#define KCOMP 32
#define DDIM  128
#define QDIM  16
// B matrix: [128 x 544]; 544 = 32*16 (V' tiles) + 2*16 (u tiles).
// Swizzled layout Bws[t][s][lane][2], t=0..33, s=0..31 (K-chunk of 4), so each
// lane issues one global_load_b64 per WMMA step, matching the B-operand VGPR map:
//   v0: lanes0-15=B[4s+0,n], lanes16-31=B[4s+2,n]; v1: rows 4s+1 / 4s+3.
#define BWS_FLOATS   (34 * 32 * 32 * 2)   // 69632
#define OFF_BVEC     69632                // 32*16
#define OFF_CK       70144                // 32
#define OFF_SQRTW    70176                // 128

// ---------------------------------------------------------------------------
// Phase A: per component k: S = I + Lam^T D^-1 Lam (16x16), Cholesky G,
// Ginv, V' = sqrt(w) .* (Lam @ Ginv^T), write B in WMMA-swizzled layout,
// plus b_k = V'^T (sqrt(w).*mu), CK_k = log_pi - 0.5*(D log2pi + logdet + c).
// ---------------------------------------------------------------------------
__global__ __launch_bounds__(256) void mfa_phaseA(
    const float* __restrict__ log_pi, const float* __restrict__ mu,
    const float* __restrict__ Lam,    const float* __restrict__ log_psi,
    float* __restrict__ Bws, float* __restrict__ bvec,
    float* __restrict__ CKws, float* __restrict__ sqrtw_ws)
{
  const int k = blockIdx.x;
  const int tid = threadIdx.x;
  __shared__ float lamS[DDIM * QDIM];   // Lam_k row-major [i][a]
  __shared__ float vS[DDIM * QDIM];     // V' [i][a]
  __shared__ float wS[DDIM];
  __shared__ float swS[DDIM];
  __shared__ float Smat[256];
  __shared__ float G[256];
  __shared__ float Gi[256];

  for (int idx = tid; idx < DDIM * QDIM; idx += 256)
    lamS[idx] = Lam[k * DDIM * QDIM + idx];
  if (tid < DDIM) {
    float d = expf(log_psi[tid]) + 1e-5f + 1e-4f;  // psi + C-jitter
    float w = 1.0f / d;
    wS[tid] = w;
    swS[tid] = sqrtf(w);
  }
  __syncthreads();

  {  // S[a][b] = delta + sum_i lam[i,a] w_i lam[i,b]   (one thread per entry)
    int a = tid >> 4, b = tid & 15;
    float s = (a == b) ? 1.0f : 0.0f;
    for (int i = 0; i < DDIM; ++i)
      s += lamS[i * 16 + a] * wS[i] * lamS[i * 16 + b];
    Smat[tid] = s;
  }
  __syncthreads();

  if (tid == 0) {  // 16x16 Cholesky + lower-triangular inverse (serial, tiny)
    for (int a = 0; a < 16; ++a) {
      float s = Smat[a * 16 + a];
      for (int m = 0; m < a; ++m) s -= G[a * 16 + m] * G[a * 16 + m];
      float ga = sqrtf(s);
      G[a * 16 + a] = ga;
      float inv = 1.0f / ga;
      for (int r = a + 1; r < 16; ++r) {
        float t = Smat[r * 16 + a];
        for (int m = 0; m < a; ++m) t -= G[r * 16 + m] * G[a * 16 + m];
        G[r * 16 + a] = t * inv;
      }
    }
    for (int a = 0; a < 16; ++a) {
      Gi[a * 16 + a] = 1.0f / G[a * 16 + a];
      for (int r = a + 1; r < 16; ++r) {
        float t = 0.0f;
        for (int m = a; m < r; ++m) t += G[r * 16 + m] * Gi[m * 16 + a];
        Gi[r * 16 + a] = -t / G[r * 16 + r];
      }
      for (int r = 0; r < a; ++r) Gi[r * 16 + a] = 0.0f;
    }
  }
  __syncthreads();

  // V'[i,a] = sqrt(w_i) * sum_{b<=a} Lam[i,b] * Ginv[a,b]; write swizzled B tile t=k
  for (int e = tid; e < DDIM * QDIM; e += 256) {
    int i = e >> 4, a = e & 15;
    float v = 0.0f;
    for (int b = 0; b <= a; ++b) v += lamS[i * 16 + b] * Gi[a * 16 + b];
    v *= swS[i];
    vS[e] = v;
    int s = i >> 2, L = ((i >> 1) & 1) * 16 + a, j = i & 1;
    Bws[((k * 32 + s) * 32 + L) * 2 + j] = v;
  }
  // u column (sqrt(w).*mu_k) into tile 32+k/16, col k%16
  if (tid < DDIM) {
    int i = tid;
    int tu = 32 + (k >> 4), cn = k & 15;
    float val = swS[i] * mu[k * DDIM + i];
    int s = i >> 2, L = ((i >> 1) & 1) * 16 + cn, j = i & 1;
    Bws[((tu * 32 + s) * 32 + L) * 2 + j] = val;
  }
  __syncthreads();

  if (tid < 16) {  // b_k[a] = sum_i V'[i,a] * sqrt(w_i) * mu_i
    float bs = 0.0f;
    for (int i = 0; i < DDIM; ++i)
      bs += vS[i * 16 + tid] * swS[i] * mu[k * DDIM + i];
    bvec[k * 16 + tid] = bs;
  }
  if (tid == 0) {
    float c = 0.0f, sumlogd = 0.0f;
    for (int i = 0; i < DDIM; ++i) {
      float m = mu[k * DDIM + i];
      c += wS[i] * m * m;
      sumlogd -= logf(wS[i]);
    }
    float logdetS = 0.0f;
    for (int a = 0; a < 16; ++a) logdetS += 2.0f * logf(G[a * 16 + a]);
    const float LOG2PI = 1.8378770664093454836f;
    CKws[k] = log_pi[k] - 0.5f * (128.0f * LOG2PI + sumlogd + logdetS + c);
  }
  if (k == 0 && tid < DDIM) sqrtw_ws[tid] = swS[tid];
}

// ---------------------------------------------------------------------------
// Phase B: fused GEMM (WMMA f32 16x16x4) + epilogue. 8 waves/block, each wave
// owns 16 sample rows; A (sqrt(w)-scaled X rows) cached in 64 VGPRs in the
// exact A-operand layout; 34 column tiles x 32 K-steps of v_wmma_f32_16x16x4_f32.
// ---------------------------------------------------------------------------
__global__ __launch_bounds__(256) void mfa_phaseB(
    const float* __restrict__ X, const float* __restrict__ Bws,
    const float* __restrict__ bvec, const float* __restrict__ CKws,
    const float* __restrict__ sqrtw,
    float* __restrict__ out_resp, float* __restrict__ out_ll, int N)
{
  __shared__ float logitS[8][16][32];
  __shared__ float swS[DDIM];
  const int tid = threadIdx.x;
  for (int i = tid; i < DDIM; i += 256) swS[i] = sqrtw[i];
  __syncthreads();

  const int wave = tid >> 5;
  const int lane = tid & 31;
  const int half = lane >> 4;   // selects K mod-4 pair per A layout
  const int ln   = lane & 15;
  const int r0   = blockIdx.x * 128 + wave * 16;
  const int row  = r0 + ln;
  const int rowc = (row < N) ? row : (N - 1);

  // Preload A' = sqrt(w) .* X[row] into registers (A-operand layout)
  v2f areg[32];
  const float* xrow = X + (size_t)rowc * DDIM;
  #pragma unroll
  for (int s = 0; s < 32; ++s) {
    int koff = 4 * s + 2 * half;
    v2f xv = *(const v2f*)(xrow + koff);
    v2f sw = *(const v2f*)(&swS[koff]);           // ds_load_b64
    areg[s].x = xv.x * sw.x;
    areg[s].y = xv.y * sw.y;
  }
  // q = sum x'^2 per row (lane pair L / L^16 hold complementary halves)
  float qs = 0.0f;
  #pragma unroll
  for (int s = 0; s < 32; ++s) qs += areg[s].x * areg[s].x + areg[s].y * areg[s].y;
  float q = qs + __shfl_xor(qs, 16, 32);

  // ---- u tiles (t=32,33): dot_{n,k} = (sqrt(w) mu_k) . x' ; seed logits ----
  #pragma unroll
  for (int ut = 0; ut < 2; ++ut) {
    const int t = 32 + ut;
    v8f acc = {};
    const float* bp = Bws + t * 2048 + lane * 2;
    __builtin_prefetch(bp, 0, 0);
    #pragma unroll
    for (int s = 0; s < 32; ++s) {
      v2f bv = *(const v2f*)(bp + s * 64);
      acc = __builtin_amdgcn_wmma_f32_16x16x4_f32(
          false, areg[s], false, bv, (short)0, acc, false, false);
    }
    const int kk = ut * 16 + ln;
    const float ckv = CKws[kk];
    #pragma unroll
    for (int r = 0; r < 8; ++r) {
      int M = r + half * 8;                 // C layout: (M=r+8*half, N=ln)
      logitS[wave][M][kk] = ckv + acc[r];
    }
  }

  // ---- V' tiles (t=k): add 0.5*||z - b_k||^2 ----
  for (int k = 0; k < KCOMP; ++k) {
    v8f acc = {};
    const float* bp = Bws + k * 2048 + lane * 2;
    __builtin_prefetch(bp + 2048, 0, 0);    // prefetch next tile
    #pragma unroll
    for (int s = 0; s < 32; ++s) {
      v2f bv = *(const v2f*)(bp + s * 64);
      acc = __builtin_amdgcn_wmma_f32_16x16x4_f32(
          false, areg[s], false, bv, (short)0, acc, false, false);
    }
    const float bval = bvec[k * 16 + ln];
    #pragma unroll
    for (int r = 0; r < 8; ++r) {
      float v = acc[r] - bval;
      float sq = v * v;
      // butterfly over the 16 lanes of each half (bit 4 untouched)
      sq += __shfl_xor(sq, 1, 32);
      sq += __shfl_xor(sq, 2, 32);
      sq += __shfl_xor(sq, 4, 32);
      sq += __shfl_xor(sq, 8, 32);
      if (ln == 0) {
        int M = r + half * 8;
        logitS[wave][M][k] += 0.5f * sq;    // same-wave LDS: in-order
      }
    }
  }

  // ---- epilogue: logsumexp over K, normalize, store ----
  if (lane < 16) {
    const int orow = r0 + lane;
    if (orow < N) {
      float l[KCOMP];
      float m = -1e30f;
      #pragma unroll
      for (int k = 0; k < KCOMP; ++k) {
        l[k] = logitS[wave][lane][k] - 0.5f * q;
        m = fmaxf(m, l[k]);
      }
      float se = 0.0f;
      #pragma unroll
      for (int k = 0; k < KCOMP; ++k) se += expf(l[k] - m);
      float lse = m + logf(se);
      #pragma unroll
      for (int k = 0; k < KCOMP; ++k)
        out_resp[(size_t)orow * KCOMP + k] = l[k] - lse;
      out_ll[orow] = lse;
    }
  }
}

extern "C" void kernel_launch(void* const* d_in, const int* in_sizes, int n_in,
                              void* d_out, int out_size, void* d_ws, size_t ws_size,
                              hipStream_t stream) {
  const float* X       = (const float*)d_in[0];
  const float* log_pi  = (const float*)d_in[1];
  const float* mu      = (const float*)d_in[2];
  const float* Lam     = (const float*)d_in[3];
  const float* log_psi = (const float*)d_in[4];
  const int N = in_sizes[0] / DDIM;

  float* ws      = (float*)d_ws;
  float* Bws     = ws;
  float* bvec    = ws + OFF_BVEC;
  float* CKws    = ws + OFF_CK;
  float* sqrtw_w = ws + OFF_SQRTW;

  float* out_resp = (float*)d_out;
  float* out_ll   = out_resp + (size_t)N * KCOMP;

  mfa_phaseA<<<KCOMP, 256, 0, stream>>>(log_pi, mu, Lam, log_psi,
                                        Bws, bvec, CKws, sqrtw_w);
  mfa_phaseB<<<(N + 127) / 128, 256, 0, stream>>>(X, Bws, bvec, CKws, sqrtw_w,
                                                  out_resp, out_ll, N);
}